// MoEGather_44762149159144
// MI455X (gfx1250) — compile-verified
//
#include <hip/hip_runtime.h>

// MoE gather + weighted top-k combine (fp32), MI455X / gfx1250.
//
// Roofline: ~192 MiB total traffic (128 MiB gathered reads, 64 MiB writes),
// ~8 us at 23.3 TB/s HBM; moe_output (128 MiB) fits in the 192 MB L2 so
// repeat-gathered rows hit L2 (~37% of gathers are repeats for random slots).
// Pure data movement: the contraction operand gathered[t,k,h] depends on both
// output indices, so there is no dense-matmul structure -> WMMA inapplicable.
// The chip-specific wins are: 128-bit coalesced vector loads (global_load_b128),
// scalar (s_load) per-token metadata, and non-temporal streaming stores so the
// 64 MiB output stream does not evict reusable rows from L2.

typedef float v4f __attribute__((ext_vector_type(4)));  // native clang vector

constexpr int kTokens = 8192;   // N_TOKENS
constexpr int kTopK   = 2;      // TOP_K (fixed by problem shape)
constexpr int kHidden = 2048;   // HIDDEN
constexpr int kCols4  = kHidden / 4;   // 512 float4 columns per row
constexpr int kBlock  = 256;           // 8 wave32 waves per block

__global__ __launch_bounds__(kBlock) void moe_gather_combine_kernel(
    const float* __restrict__ moe_output,    // [kTokens*kTopK, kHidden]
    const float* __restrict__ scores,        // [kTokens*kTopK]
    const int*   __restrict__ mapped_slots,  // [kTokens*kTopK]
    float* __restrict__ out)                 // [kTokens, kHidden]
{
    const int t = blockIdx.x;  // one token per block (block-uniform)

    // Block-uniform metadata: compiler should lower these to s_load_* (SMEM).
    const int   s0 = mapped_slots[kTopK * t + 0];
    const int   s1 = mapped_slots[kTopK * t + 1];
    const float w0 = scores[kTopK * t + 0];
    const float w1 = scores[kTopK * t + 1];

    const v4f* __restrict__ r0 =
        reinterpret_cast<const v4f*>(moe_output) + (size_t)s0 * kCols4;
    const v4f* __restrict__ r1 =
        reinterpret_cast<const v4f*>(moe_output) + (size_t)s1 * kCols4;
    v4f* __restrict__ o =
        reinterpret_cast<v4f*>(out) + (size_t)t * kCols4;

#pragma unroll
    for (int base = 0; base < kCols4; base += kBlock) {
        const int c = base + (int)threadIdx.x;
        // Temporal (RT) loads: gathered rows have expected reuse and the whole
        // moe_output tensor fits in the 192 MB global L2.
        const v4f a = r0[c];
        const v4f b = r1[c];
        v4f v;
        v.x = fmaf(w0, a.x, w1 * b.x);
        v.y = fmaf(w0, a.y, w1 * b.y);
        v.z = fmaf(w0, a.z, w1 * b.z);
        v.w = fmaf(w0, a.w, w1 * b.w);
        // Streaming output: non-temporal store (NT hint) for the write stream.
        __builtin_nontemporal_store(v, o + c);
    }
}

extern "C" void kernel_launch(void* const* d_in, const int* in_sizes, int n_in,
                              void* d_out, int out_size, void* d_ws, size_t ws_size,
                              hipStream_t stream) {
    (void)in_sizes; (void)n_in; (void)out_size; (void)d_ws; (void)ws_size;
    const float* moe_output   = (const float*)d_in[0];
    const float* scores       = (const float*)d_in[1];
    const int*   mapped_slots = (const int*)d_in[2];
    // d_in[3] = top_k scalar; fixed at 2 by the reference shapes.
    float* out = (float*)d_out;

    moe_gather_combine_kernel<<<dim3(kTokens), dim3(kBlock), 0, stream>>>(
        moe_output, scores, mapped_slots, out);
}